// ArchitectureBlock_36807869727136
// MI455X (gfx1250) — compile-verified
//
#include <hip/hip_runtime.h>
#include <hip/hip_bf16.h>

typedef float v2f __attribute__((ext_vector_type(2)));
typedef float v8f __attribute__((ext_vector_type(8)));
typedef int async_v4i __attribute__((vector_size(16)));   // matches builtin param

// ---------------------------------------------------------------------------
// CDNA5 async global->LDS staging helpers (ASYNCcnt path), with safe fallback.
// ---------------------------------------------------------------------------
__device__ __forceinline__ void async_copy16(const float* g, float* l) {
#if __has_builtin(__builtin_amdgcn_global_load_async_to_lds_b128)
    __builtin_amdgcn_global_load_async_to_lds_b128(
        (__attribute__((address_space(1))) async_v4i*)g,
        (__attribute__((address_space(3))) async_v4i*)l, 0, 0);
#else
    *(float4*)l = *(const float4*)g;
#endif
}

__device__ __forceinline__ void async_wait_all() {
#if __has_builtin(__builtin_amdgcn_global_load_async_to_lds_b128)
#if __has_builtin(__builtin_amdgcn_s_wait_asynccnt)
    __builtin_amdgcn_s_wait_asynccnt(0);
#else
    asm volatile("s_wait_asynccnt 0x0" ::: "memory");
#endif
#endif
}

// ---------------------------------------------------------------------------
// LayerNorm: one block (256 threads) per row of length D.
// ---------------------------------------------------------------------------
__global__ __launch_bounds__(256) void ln_kernel(const float* __restrict__ x,
                                                 const float* __restrict__ gamma,
                                                 const float* __restrict__ beta,
                                                 float* __restrict__ out, int D) {
    const int row = blockIdx.x;
    const int tid = threadIdx.x;
    const float* xr = x + (size_t)row * D;
    float* orow = out + (size_t)row * D;

    float sum = 0.f, sq = 0.f;
    for (int i = tid; i < (D >> 2); i += 256) {
        float4 v = *(const float4*)(xr + 4 * i);
        sum += v.x + v.y + v.z + v.w;
        sq  += v.x * v.x + v.y * v.y + v.z * v.z + v.w * v.w;
    }
    for (int off = 16; off > 0; off >>= 1) {
        sum += __shfl_down(sum, off, 32);
        sq  += __shfl_down(sq,  off, 32);
    }
    __shared__ float red[2][8];
    __shared__ float stats[2];
    const int wave = tid >> 5, lane = tid & 31;
    if (lane == 0) { red[0][wave] = sum; red[1][wave] = sq; }
    __syncthreads();
    if (tid == 0) {
        float s = 0.f, q = 0.f;
        for (int w = 0; w < 8; ++w) { s += red[0][w]; q += red[1][w]; }
        float mean = s / (float)D;
        float var  = q / (float)D - mean * mean;
        stats[0] = mean;
        stats[1] = rsqrtf(var + 1e-5f);
    }
    __syncthreads();
    const float mean = stats[0], rstd = stats[1];
    for (int i = tid; i < (D >> 2); i += 256) {
        float4 v = *(const float4*)(xr + 4 * i);
        float4 g = *(const float4*)(gamma + 4 * i);
        float4 b = *(const float4*)(beta + 4 * i);
        float4 o;
        o.x = (v.x - mean) * rstd * g.x + b.x;
        o.y = (v.y - mean) * rstd * g.y + b.y;
        o.z = (v.z - mean) * rstd * g.z + b.z;
        o.w = (v.w - mean) * rstd * g.w + b.w;
        *(float4*)(orow + 4 * i) = o;
    }
}

// ---------------------------------------------------------------------------
// Spiral conv as first-order complex recurrence (replaces the FFT):
//   state[l] = phazor*state[l-1] + xn[l]
//   cwp[l]   = phazor_init*state[l] + hidden*phazor^{l+1}
// ---------------------------------------------------------------------------
__global__ __launch_bounds__(256) void scan_kernel(
    const float* __restrict__ xn, const float* __restrict__ hr,
    const float* __restrict__ hi, const float* __restrict__ pr,
    const float* __restrict__ pi, const float* __restrict__ pir,
    const float* __restrict__ pii, float* __restrict__ s_out,
    float* __restrict__ nh_out, int B, int L, int D) {
    const int idx = blockIdx.x * blockDim.x + threadIdx.x;
    if (idx >= B * D) return;
    const int d = idx % D, b = idx / D;

    const float ar = pr[d], ai = pi[d];
    const float amag = sqrtf(ar * ar + ai * ai);
    const float scl = __expf(-amag) / amag;   // p/|p| * exp(-|p|)
    const float phr = ar * scl, phi = ai * scl;
    const float ir = pir[d], ii = pii[d];

    float str = 0.f, sti = 0.f;
    const float h0r = hr[(size_t)b * D + d], h0i = hi[(size_t)b * D + d];
    float tr = h0r * phr - h0i * phi;
    float ti = h0r * phi + h0i * phr;

    for (int l = 0; l < L; ++l) {
        const size_t o = ((size_t)b * L + l) * D + d;
        const float xv = xn[o];
        const float nr = phr * str - phi * sti + xv;
        const float ni = phr * sti + phi * str;
        str = nr; sti = ni;
        const float cr = ir * str - ii * sti + tr;
        const float ci = ir * sti + ii * str + ti;
        s_out[o] = cr;
        nh_out[2 * o]     = cr;
        nh_out[2 * o + 1] = ci;
        const float t2r = tr * phr - ti * phi;
        const float t2i = tr * phi + ti * phr;
        tr = t2r; ti = t2i;
    }
}

// ---------------------------------------------------------------------------
// WMMA f32 GEMM:  C[m][n] = sum_k A[m][k] * W[n][k]   (W row-major [N,K])
// 256 thr = 8 waves; block tile 128x64; wave tile 16x64 (4 v8f accs).
// Double-buffered LDS tiles filled by async global->LDS copies so the next
// K-tile's fetch overlaps this tile's v_wmma work.
// MODE 0: out = aux1 * silu(C + bias) + aux2      (fc + s*y + x_res)
// MODE 1: out = silu(C + bias)                    (ffn first layer)
// MODE 2: out = C + bias + aux2                   (ffn second layer + res)
// ---------------------------------------------------------------------------
#define LSTR 20

__device__ __forceinline__ void stage_tiles(const float* __restrict__ A,
                                            const float* __restrict__ W,
                                            float (*As)[LSTR], float (*Bs)[LSTR],
                                            int mBase, int nBase, int kt, int K,
                                            int tid) {
    const int row = tid >> 2, c4 = (tid & 3) * 4;
    async_copy16(A + (size_t)(mBase + row) * K + kt + c4, &As[row][c4]);
    async_copy16(A + (size_t)(mBase + 64 + row) * K + kt + c4, &As[64 + row][c4]);
    async_copy16(W + (size_t)(nBase + row) * K + kt + c4, &Bs[row][c4]);
}

template <int MODE>
__global__ __launch_bounds__(256, 2) void gemm_wmma_kernel(
    const float* __restrict__ A, const float* __restrict__ W,
    const float* __restrict__ bias, const float* __restrict__ aux1,
    const float* __restrict__ aux2, float* __restrict__ out,
    int M, int N, int K) {
    __shared__ float As[2][128][LSTR];
    __shared__ float Bs[2][64][LSTR];

    const int tid  = threadIdx.x;
    const int wave = tid >> 5;
    const int lane = tid & 31;
    const int mBase = blockIdx.y * 128;
    const int nBase = blockIdx.x * 64;
    const int mOff  = wave * 16;
    const int nl   = lane & 15;
    const int half = lane >> 4;

    v8f acc[4];
#pragma unroll
    for (int t = 0; t < 4; ++t)
#pragma unroll
        for (int r = 0; r < 8; ++r) acc[t][r] = 0.f;

    // prologue: stage first K-tile
    stage_tiles(A, W, As[0], Bs[0], mBase, nBase, 0, K, tid);
    async_wait_all();
    __syncthreads();

    int buf = 0;
    for (int kt = 0; kt < K; kt += 16) {
        // prefetch next tile into the other buffer (overlaps with WMMA below)
        if (kt + 16 < K)
            stage_tiles(A, W, As[buf ^ 1], Bs[buf ^ 1], mBase, nBase, kt + 16, K, tid);

#pragma unroll
        for (int kk = 0; kk < 16; kk += 4) {
            const v2f av = *(const v2f*)&As[buf][mOff + nl][kk + 2 * half];
#pragma unroll
            for (int t = 0; t < 4; ++t) {
                const v2f bv = *(const v2f*)&Bs[buf][t * 16 + nl][kk + 2 * half];
                acc[t] = __builtin_amdgcn_wmma_f32_16x16x4_f32(
                    false, av, false, bv, (short)0, acc[t], false, false);
            }
        }

        async_wait_all();   // next tile fully landed in LDS
        __syncthreads();    // all waves done reading current tile
        buf ^= 1;
    }

    // --- fused epilogue ---
#pragma unroll
    for (int t = 0; t < 4; ++t) {
        const int n = nBase + t * 16 + nl;
        const float bn = bias[n];
#pragma unroll
        for (int r = 0; r < 8; ++r) {
            const int m = mBase + mOff + half * 8 + r;
            const size_t idx = (size_t)m * N + n;
            float v = acc[t][r] + bn;
            if (MODE == 0) {
                const float y = v * (1.f / (1.f + __expf(-v)));   // silu
                out[idx] = aux1[idx] * y + aux2[idx];
            } else if (MODE == 1) {
                out[idx] = v * (1.f / (1.f + __expf(-v)));
            } else {
                out[idx] = v + aux2[idx];
            }
        }
    }
}

// ---------------------------------------------------------------------------
extern "C" void kernel_launch(void* const* d_in, const int* in_sizes, int n_in,
                              void* d_out, int out_size, void* d_ws, size_t ws_size,
                              hipStream_t stream) {
    const float* x    = (const float*)d_in[0];
    const float* hr   = (const float*)d_in[1];
    const float* hi   = (const float*)d_in[2];
    const float* pr   = (const float*)d_in[3];
    const float* pi   = (const float*)d_in[4];
    const float* pir  = (const float*)d_in[5];
    const float* pii  = (const float*)d_in[6];
    const float* lng  = (const float*)d_in[7];
    const float* lnb  = (const float*)d_in[8];
    const float* fc_w = (const float*)d_in[9];
    const float* fc_b = (const float*)d_in[10];
    const float* w1   = (const float*)d_in[11];
    const float* b1   = (const float*)d_in[12];
    const float* w2   = (const float*)d_in[13];
    const float* b2   = (const float*)d_in[14];

    const int D  = in_sizes[3];                       // 1024
    const int B  = in_sizes[1] / D;                   // 4
    const int L  = in_sizes[0] / (B * D);             // 2048
    const int DF = in_sizes[12];                      // 4096
    const int M  = B * L;                             // 8192

    float* ws = (float*)d_ws;
    float* xn = ws;                                   // M*D  (later reused as hn)
    float* s  = xn + (size_t)M * D;                   // M*D
    float* h  = s  + (size_t)M * D;                   // M*D
    float* u  = h  + (size_t)M * D;                   // M*DF

    float* out = (float*)d_out;                       // M*D
    float* nh  = out + (size_t)M * D;                 // M*D complex, interleaved

    // 1) xn = LN(x)
    ln_kernel<<<M, 256, 0, stream>>>(x, lng, lnb, xn, D);

    // 2) spiral-conv scan -> s (real part) + new_hidden (complex, d_out tail)
    scan_kernel<<<(B * D + 255) / 256, 256, 0, stream>>>(
        xn, hr, hi, pr, pi, pir, pii, s, nh, B, L, D);

    // 3) h = s * silu(x @ fc_w^T + fc_b) + x
    {
        dim3 grid(D / 64, M / 128);
        gemm_wmma_kernel<0><<<grid, 256, 0, stream>>>(x, fc_w, fc_b, s, x, h, M, D, D);
    }

    // 4) hn = LN(h)   (reuse xn buffer)
    ln_kernel<<<M, 256, 0, stream>>>(h, lng, lnb, xn, D);

    // 5) u = silu(hn @ w1^T + b1)
    {
        dim3 grid(DF / 64, M / 128);
        gemm_wmma_kernel<1><<<grid, 256, 0, stream>>>(xn, w1, b1, nullptr, nullptr, u, M, DF, D);
    }

    // 6) out = u @ w2^T + b2 + h
    {
        dim3 grid(D / 64, M / 128);
        gemm_wmma_kernel<2><<<grid, 256, 0, stream>>>(u, w2, b2, nullptr, h, out, M, D, DF);
    }
}